// MyCoAttention_21552145891847
// MI455X (gfx1250) — compile-verified
//
#include <hip/hip_runtime.h>
#include <cmath>

#define B_  8
#define TE_ 256
#define TD_ 256
#define D_  256
#define U_  256

typedef float v2f __attribute__((ext_vector_type(2)));
typedef float v8f __attribute__((ext_vector_type(8)));

// ---------------------------------------------------------------------------
// C[M x 256] = A[M x 256] * W[256 x 256], fp32, V_WMMA_F32_16X16X4_F32.
// One 16x16 C tile per wave, 8 waves per 256-thread block.
// A fragment: lane -> (m = lane&15, khalf = lane>>4), VGPRs = K 2*khalf+{0,1}
// B fragment: lane -> (n = lane&15), rows K = 2*khalf+{0,1}
// C fragment: VGPR r -> (m = r + 8*khalf, n = lane&15)
// ---------------------------------------------------------------------------
__global__ void __launch_bounds__(256) gemm_f32_wmma(
    const float* __restrict__ A, const float* __restrict__ W,
    float* __restrict__ C) {
  const int lane  = threadIdx.x & 31;
  const int wave  = threadIdx.x >> 5;
  const int tile  = blockIdx.x * 8 + wave;
  const int tn    = tile & 15;           // 256/16 = 16 tiles along N
  const int tm    = tile >> 4;
  const int m     = lane & 15;
  const int khalf = lane >> 4;

  const float* arow = A + (size_t)(tm * 16 + m) * U_;
  const float* bcol = W + tn * 16 + m;   // n == lane&15
  v8f acc = {};
  for (int k = 0; k < U_; k += 4) {
    const int ka = k + 2 * khalf;
    v2f a, b;
    a.x = arow[ka + 0];
    a.y = arow[ka + 1];
    b.x = bcol[(size_t)(ka + 0) * U_];
    b.y = bcol[(size_t)(ka + 1) * U_];
    acc = __builtin_amdgcn_wmma_f32_16x16x4_f32(false, a, false, b,
                                                (short)0, acc, false, false);
  }
#pragma unroll
  for (int r = 0; r < 8; ++r) {
    const int m_out = tm * 16 + r + 8 * khalf;
    C[(size_t)m_out * U_ + tn * 16 + m] = acc[r];
  }
}

// ---------------------------------------------------------------------------
// mu[b,s,t] = sum_u tanh(ad[b,s,u] + ae[b,t,u]) * nu[u]
// 16(s) x 16(t) tile per block; ad/ae tiles staged in LDS with pitch 260
// (260 % 64 = 4 -> the 16 lanes hitting column u of aeT land on distinct banks).
// This is the fused form of the 537MB "co" tensor: it never touches HBM.
// ---------------------------------------------------------------------------
__global__ void __launch_bounds__(256) co_mu_kernel(
    const float* __restrict__ ae, const float* __restrict__ ad,
    const float* __restrict__ nu, float* __restrict__ mu) {
  const int b      = blockIdx.y;
  const int s_tile = blockIdx.x >> 4;    // TD/16 = 16
  const int t_tile = blockIdx.x & 15;    // TE/16 = 16
  __shared__ float adT[16][260];
  __shared__ float aeT[16][260];
  __shared__ float nuS[256];

  const int tid = threadIdx.x;
  nuS[tid] = nu[tid];
  for (int i = tid; i < 16 * 256; i += 256) {
    const int r = i >> 8, c = i & 255;
    adT[r][c] = ad[((size_t)(b * TD_ + s_tile * 16 + r)) * U_ + c];
    aeT[r][c] = ae[((size_t)(b * TE_ + t_tile * 16 + r)) * U_ + c];
  }
  __syncthreads();

  const int si = tid >> 4, ti = tid & 15;
  float acc = 0.f;
#pragma unroll 4
  for (int u = 0; u < U_; ++u)
    acc += tanhf(adT[si][u] + aeT[ti][u]) * nuS[u];

  mu[((size_t)(b * TD_ + s_tile * 16 + si)) * TE_ + t_tile * 16 + ti] = acc;
}

// ---------------------------------------------------------------------------
// In-place softmax over the TE axis; also emits the row max (== max_mu).
// One 256-thread block per (b, s) row.
// ---------------------------------------------------------------------------
__global__ void __launch_bounds__(256) softmax_rows_kernel(
    float* __restrict__ mu, float* __restrict__ max_mu) {
  const int row = blockIdx.x;            // b*TD + s
  const int tid = threadIdx.x;
  __shared__ float red[256];

  const float v = mu[(size_t)row * TE_ + tid];
  red[tid] = v; __syncthreads();
  for (int s = 128; s > 0; s >>= 1) {
    if (tid < s) red[tid] = fmaxf(red[tid], red[tid + s]);
    __syncthreads();
  }
  const float mx = red[0];
  __syncthreads();
  const float e = expf(v - mx);
  red[tid] = e; __syncthreads();
  for (int s = 128; s > 0; s >>= 1) {
    if (tid < s) red[tid] += red[tid + s];
    __syncthreads();
  }
  mu[(size_t)row * TE_ + tid] = e * (1.0f / red[0]);
  if (tid == 0) max_mu[row] = mx;
}

// ---------------------------------------------------------------------------
// max_alphas = softmax_s(max_mu[b,:]);  h_hat[b,d] = sum_s de[b,s,d]*max_alphas[s]
// One block per batch.
// ---------------------------------------------------------------------------
__global__ void __launch_bounds__(256) hhat_kernel(
    const float* __restrict__ max_mu, const float* __restrict__ de_seq,
    float* __restrict__ h_hat) {
  const int b   = blockIdx.x;
  const int tid = threadIdx.x;
  __shared__ float red[256];
  __shared__ float ma[256];

  const float v = max_mu[b * TD_ + tid];
  red[tid] = v; __syncthreads();
  for (int s = 128; s > 0; s >>= 1) {
    if (tid < s) red[tid] = fmaxf(red[tid], red[tid + s]);
    __syncthreads();
  }
  const float mx = red[0];
  __syncthreads();
  const float e = expf(v - mx);
  red[tid] = e; __syncthreads();
  for (int s = 128; s > 0; s >>= 1) {
    if (tid < s) red[tid] += red[tid + s];
    __syncthreads();
  }
  ma[tid] = e * (1.0f / red[0]);
  __syncthreads();

  float acc = 0.f;
  for (int s = 0; s < TD_; ++s)
    acc += de_seq[((size_t)(b * TD_ + s)) * D_ + tid] * ma[s];
  h_hat[b * D_ + tid] = acc;
}

// ---------------------------------------------------------------------------
// sum_en[b] = alphas[b] (TDxTE) @ en_seq[b] (TExD) via fp32 WMMA, with the
// concat epilogue fused straight out of the accumulator fragment:
// out = [de, sum_en, de*sum_en, de*h_hat]
// ---------------------------------------------------------------------------
__global__ void __launch_bounds__(256) sum_en_out_kernel(
    const float* __restrict__ alphas, const float* __restrict__ en_seq,
    const float* __restrict__ de_seq, const float* __restrict__ h_hat,
    float* __restrict__ out) {
  const int b     = blockIdx.y;
  const int lane  = threadIdx.x & 31;
  const int wave  = threadIdx.x >> 5;
  const int tile  = blockIdx.x * 8 + wave;  // 256 tiles/batch, gridDim.x = 32
  const int td    = tile & 15;              // D/16
  const int ts    = tile >> 4;              // TD/16
  const int m     = lane & 15;
  const int khalf = lane >> 4;

  const float* A    = alphas + (size_t)b * TD_ * TE_;
  const float* Bm   = en_seq + (size_t)b * TE_ * D_;
  const float* arow = A + (size_t)(ts * 16 + m) * TE_;
  const float* bcol = Bm + td * 16 + m;

  v8f acc = {};
  for (int k = 0; k < TE_; k += 4) {
    const int ka = k + 2 * khalf;
    v2f a, bb;
    a.x  = arow[ka + 0];
    a.y  = arow[ka + 1];
    bb.x = bcol[(size_t)(ka + 0) * D_];
    bb.y = bcol[(size_t)(ka + 1) * D_];
    acc = __builtin_amdgcn_wmma_f32_16x16x4_f32(false, a, false, bb,
                                                (short)0, acc, false, false);
  }

  const int d_g  = td * 16 + m;
  const float hh = h_hat[b * D_ + d_g];
#pragma unroll
  for (int r = 0; r < 8; ++r) {
    const int s_g  = ts * 16 + r + 8 * khalf;
    const float se = acc[r];
    const float de = de_seq[((size_t)(b * TD_ + s_g)) * D_ + d_g];
    float* o = out + ((size_t)(b * TD_ + s_g)) * (4 * D_);
    o[d_g]          = de;
    o[D_ + d_g]     = se;
    o[2 * D_ + d_g] = de * se;
    o[3 * D_ + d_g] = de * hh;
  }
}

extern "C" void kernel_launch(void* const* d_in, const int* in_sizes, int n_in,
                              void* d_out, int out_size, void* d_ws, size_t ws_size,
                              hipStream_t stream) {
  const float* en_seq = (const float*)d_in[0];
  const float* de_seq = (const float*)d_in[1];
  const float* w_en   = (const float*)d_in[2];
  const float* w_de   = (const float*)d_in[3];
  const float* nu     = (const float*)d_in[4];
  float* out = (float*)d_out;

  // Workspace layout (floats): ae | ad | mu(->alphas in place) | max_mu | h_hat
  float* ws = (float*)d_ws;
  const size_t nAE = (size_t)B_ * TE_ * U_;       // 524288
  float* ae     = ws;
  float* ad     = ae + nAE;
  float* mu     = ad + nAE;                       // B*TD*TE
  float* max_mu = mu + (size_t)B_ * TD_ * TE_;    // B*TD
  float* h_hat  = max_mu + (size_t)B_ * TD_;      // B*D

  const dim3 blk(256);
  // 2048 rows -> 2048 tiles -> 256 blocks of 8 waves
  gemm_f32_wmma<<<dim3(256), blk, 0, stream>>>(en_seq, w_en, ae);
  gemm_f32_wmma<<<dim3(256), blk, 0, stream>>>(de_seq, w_de, ad);
  co_mu_kernel<<<dim3(256, B_), blk, 0, stream>>>(ae, ad, nu, mu);
  softmax_rows_kernel<<<dim3(B_ * TD_), blk, 0, stream>>>(mu, max_mu);
  hhat_kernel<<<dim3(B_), blk, 0, stream>>>(max_mu, de_seq, h_hat);
  sum_en_out_kernel<<<dim3(32, B_), blk, 0, stream>>>(mu, en_seq, de_seq,
                                                      h_hat, out);
}